// GCNConvGNNLayer_34772055229050
// MI455X (gfx1250) — compile-verified
//
#include <hip/hip_runtime.h>
#include <hip/hip_bf16.h>

typedef __attribute__((ext_vector_type(2))) float v2f;
typedef __attribute__((ext_vector_type(8))) float v8f;

#define DIM 128
#define LDS_STRIDE 132  // 128 + 4 pad: bank(c + 4*row) distinct for 16 rows

// ---------------------------------------------------------------------------
// 1) zero agg (=d_out) and deg
// ---------------------------------------------------------------------------
__global__ __launch_bounds__(256) void zero_kernel(float* __restrict__ agg, size_t n_agg,
                                                   float* __restrict__ deg, int n_deg) {
    size_t i = (size_t)blockIdx.x * blockDim.x + threadIdx.x;
    size_t stride = (size_t)gridDim.x * blockDim.x;
    for (size_t j = i; j < n_agg; j += stride) agg[j] = 0.0f;
    for (size_t j = i; j < (size_t)n_deg; j += stride) deg[j] = 0.0f;
}

// ---------------------------------------------------------------------------
// 2) in-degree via f32 atomics (exact for counts < 2^24)
//    dst[] is streamed exactly once -> non-temporal load keeps L2 for h/agg
// ---------------------------------------------------------------------------
__global__ __launch_bounds__(256) void deg_kernel(const int* __restrict__ dst,
                                                  float* __restrict__ deg, int E) {
    int i = blockIdx.x * blockDim.x + threadIdx.x;
    if (i < E) {
        int d = __builtin_nontemporal_load(dst + i);
        atomicAdd(&deg[d], 1.0f);
    }
}

// ---------------------------------------------------------------------------
// 3) deg -> deg^{-1/2} in place (self-loop adds 1, so deg+1 >= 1 always)
// ---------------------------------------------------------------------------
__global__ __launch_bounds__(256) void dinv_kernel(float* __restrict__ deg, int N) {
    int i = blockIdx.x * blockDim.x + threadIdx.x;
    if (i < N) deg[i] = rsqrtf(deg[i] + 1.0f);
}

// ---------------------------------------------------------------------------
// 4) h = x @ W with V_WMMA_F32_16X16X4_F32 (fp32 in/out, K-step 4)
//    block = 256 threads (8 wave32), one 16-row M-tile; wave w owns N-tile w.
// ---------------------------------------------------------------------------
__global__ __launch_bounds__(256) void gemm_wmma_kernel(const float* __restrict__ x,
                                                        const float* __restrict__ W,
                                                        float* __restrict__ h, int N) {
    __shared__ float lds_a[16 * LDS_STRIDE];

    const int row0 = blockIdx.x * 16;
    const int t = threadIdx.x;
    const bool full_tile = (row0 + 16 <= N);

    // cooperative coalesced load of the 16x128 x-tile into LDS (float4)
    for (int i = t; i < (16 * DIM) / 4; i += 256) {
        int r = (i * 4) / DIM;
        int c = (i * 4) % DIM;
        int row = row0 + r;
        float4 v = make_float4(0.f, 0.f, 0.f, 0.f);
        if (full_tile || row < N) v = *(const float4*)(x + (size_t)row * DIM + c);
        float* p = &lds_a[r * LDS_STRIDE + c];
        p[0] = v.x; p[1] = v.y; p[2] = v.z; p[3] = v.w;
    }
    __syncthreads();

    const int wave = t >> 5;          // 0..7 -> which 16-col N-tile
    const int lane = t & 31;
    const int mn   = lane & 15;       // A: row M; B: col N (within tile)
    const int kk   = (lane >> 4) * 2; // lanes 0-15: K=0,1 ; lanes 16-31: K=2,3
    const int ncol = wave * 16 + mn;

    v8f acc = {};
    #pragma unroll 4
    for (int k = 0; k < DIM; k += 4) {
        v2f a, bf;
        a.x  = lds_a[mn * LDS_STRIDE + k + kk];
        a.y  = lds_a[mn * LDS_STRIDE + k + kk + 1];
        bf.x = W[(size_t)(k + kk)     * DIM + ncol];   // L2/WGP$-resident (64 KB)
        bf.y = W[(size_t)(k + kk + 1) * DIM + ncol];
        // (neg_a, A, neg_b, B, c_mod, C, reuse_a, reuse_b)
        acc = __builtin_amdgcn_wmma_f32_16x16x4_f32(false, a, false, bf,
                                                    (short)0, acc, false, false);
    }

    // D layout: VGPR r -> M=r (lanes 0-15) / M=r+8 (lanes 16-31), N = lane&15
    const int rbase = row0 + ((lane >> 4) * 8);
    float* hp = h + (size_t)rbase * DIM + ncol;
    if (full_tile) {
        #pragma unroll
        for (int r = 0; r < 8; ++r) hp[(size_t)r * DIM] = acc[r];
    } else {
        #pragma unroll
        for (int r = 0; r < 8; ++r)
            if (rbase + r < N) hp[(size_t)r * DIM] = acc[r];
    }
}

// ---------------------------------------------------------------------------
// 5) edge scatter: one wave per edge, 32 lanes x float4 = 128 floats
//    src/dst streamed once (non-temporal); h rows reused ~17x (keep cached)
// ---------------------------------------------------------------------------
__global__ __launch_bounds__(256) void edge_scatter_kernel(const float* __restrict__ h,
                                                           const int* __restrict__ src,
                                                           const int* __restrict__ dst,
                                                           const float* __restrict__ dinv,
                                                           float* __restrict__ agg, int E) {
    const int wave = blockIdx.x * (blockDim.x >> 5) + (threadIdx.x >> 5);
    const int lane = threadIdx.x & 31;
    if (wave >= E) return;

    const int s = __builtin_nontemporal_load(src + wave);
    const int d = __builtin_nontemporal_load(dst + wave);
    const float w = dinv[s] * dinv[d];

    const float4 v = *(const float4*)(h + (size_t)s * DIM + lane * 4);
    float* o = agg + (size_t)d * DIM + lane * 4;
    atomicAdd(o + 0, v.x * w);
    atomicAdd(o + 1, v.y * w);
    atomicAdd(o + 2, v.z * w);
    atomicAdd(o + 3, v.w * w);
}

// ---------------------------------------------------------------------------
// 6) out = x + relu(agg + h*dinv^2 + b)   (self-loop folded; in-place on agg)
// ---------------------------------------------------------------------------
__global__ __launch_bounds__(256) void final_kernel(const float* __restrict__ x,
                                                    const float* __restrict__ h,
                                                    const float* __restrict__ dinv,
                                                    const float* __restrict__ b,
                                                    float* __restrict__ out, int N) {
    size_t q = (size_t)blockIdx.x * blockDim.x + threadIdx.x;  // float4 index
    size_t nq = (size_t)N * (DIM / 4);
    if (q >= nq) return;

    size_t base = q * 4;
    int node = (int)(base / DIM);
    int col  = (int)(base % DIM);

    float w = dinv[node];
    w = w * w;

    float4 hv = *(const float4*)(h + base);
    float4 av = *(const float4*)(out + base);
    float4 xv = *(const float4*)(x + base);
    float4 bv = *(const float4*)(b + col);

    float4 r;
    r.x = xv.x + fmaxf(av.x + hv.x * w + bv.x, 0.0f);
    r.y = xv.y + fmaxf(av.y + hv.y * w + bv.y, 0.0f);
    r.z = xv.z + fmaxf(av.z + hv.z * w + bv.z, 0.0f);
    r.w = xv.w + fmaxf(av.w + hv.w * w + bv.w, 0.0f);
    *(float4*)(out + base) = r;
}

// ---------------------------------------------------------------------------
extern "C" void kernel_launch(void* const* d_in, const int* in_sizes, int n_in,
                              void* d_out, int out_size, void* d_ws, size_t ws_size,
                              hipStream_t stream) {
    const float* x  = (const float*)d_in[0];
    const float* W  = (const float*)d_in[1];
    const float* b  = (const float*)d_in[2];
    const int*   ei = (const int*)d_in[3];

    const int N = in_sizes[0] / DIM;
    const int E = in_sizes[3] / 2;
    const int* src = ei;       // edge_index[0]
    const int* dst = ei + E;   // edge_index[1]

    float* h   = (float*)d_ws;                 // N*DIM floats
    float* deg = h + (size_t)N * DIM;          // N floats (becomes dinv in place)
    float* agg = (float*)d_out;                // accumulate directly into output

    const size_t nd = (size_t)N * DIM;

    zero_kernel<<<2048, 256, 0, stream>>>(agg, nd, deg, N);
    deg_kernel<<<(E + 255) / 256, 256, 0, stream>>>(dst, deg, E);
    dinv_kernel<<<(N + 255) / 256, 256, 0, stream>>>(deg, N);

    gemm_wmma_kernel<<<(N + 15) / 16, 256, 0, stream>>>(x, W, h, N);

    edge_scatter_kernel<<<(E + 7) / 8, 256, 0, stream>>>(h, src, dst, deg, agg, E);

    const size_t nq = nd / 4;
    final_kernel<<<(int)((nq + 255) / 256), 256, 0, stream>>>(x, h, deg, b, agg, N);
}